// LiftSplatShoot_10144712753203
// MI455X (gfx1250) — compile-verified
//
#include <hip/hip_runtime.h>

// ---- types for WMMA ----
typedef float v2f  __attribute__((ext_vector_type(2)));
typedef float v8f  __attribute__((ext_vector_type(8)));

#define BB   8
#define NNC  6
#define CC   64
#define DDp  41
#define HH   8
#define WW   22
#define NXY  200
#define CSTRIDE   (NXY*NXY)        /* 40000 */
#define OUT_PER_B (CC*CSTRIDE)     /* 2,560,000 */

// Exact f32 outer-product tile on the matrix pipe:
//   A: 16x4, only K=0 column nonzero (lanes 0-15 / first VGPR hold M=0..15,K=0)
//   B: 4x16, only K=0 row nonzero    (lanes 0-15 / first VGPR hold K=0,N=0..15)
//   D[m][n] = aval_m * bval_n   (single f32 multiply, RNE -> exact)
static __device__ inline v8f outer_wmma(float aval, float bval, int lane) {
  v2f A; A.x = (lane < 16) ? aval : 0.0f; A.y = 0.0f;
  v2f B; B.x = (lane < 16) ? bval : 0.0f; B.y = 0.0f;
  v8f acc = {};
  return __builtin_amdgcn_wmma_f32_16x16x4_f32(false, A, false, B, (short)0, acc,
                                               false, false);
}

__global__ __launch_bounds__(256) void lss_zero_f4(float4* __restrict__ out, int n4) {
  int i = blockIdx.x * blockDim.x + threadIdx.x;
  if (i < n4) out[i] = float4{0.0f, 0.0f, 0.0f, 0.0f};
}

__global__ __launch_bounds__(256) void lss_lift_splat(
    const float* __restrict__ depth_logits, // (B,N,41,8,22)
    const float* __restrict__ feats,        // (B,N,64,8,22)
    const float* __restrict__ rots,         // (B,N,3,3)
    const float* __restrict__ trans,        // (B,N,3)
    const float* __restrict__ Kc,           // (B,N,4) g1,g2,u0,v0
    const float* __restrict__ Dc,           // (B,N,4) k1,k2,p1,p2
    const float* __restrict__ xiP,          // (B,N)
    float* __restrict__ out)                // (B,64,200,200)
{
  const int lane = threadIdx.x & 31;
  const int wid  = threadIdx.x >> 5;
  const int pix  = blockIdx.x * 8 + wid;              // one wave per pixel
  if (pix >= BB * NNC * HH * WW) return;
  const int w = pix % WW;
  const int h = (pix / WW) % HH;
  const int n = (pix / (WW * HH)) % NNC;
  const int b = pix / (WW * HH * NNC);
  const int cam = b * NNC + n;

  // ---- camera intrinsics + mask (wave-uniform early exit) ----
  const float g1 = Kc[cam*4+0], g2 = Kc[cam*4+1];
  const float u0 = Kc[cam*4+2], v0 = Kc[cam*4+3];
  const float u0m = floorf(u0 * (1.0f/16.0f));
  const float v0m = floorf(v0 * (1.0f/16.0f));
  const float mx = (float)w - u0m, my = (float)h - v0m;
  if (mx*mx + my*my >= 9.0f) return;  // masked pixel contributes exactly 0

  const float dk1 = Dc[cam*4+0], dk2 = Dc[cam*4+1];
  const float dp1 = Dc[cam*4+2], dp2 = Dc[cam*4+3];
  const float xiv = xiP[cam];

  // ---- undistortion (20 fixed-point iterations) ----
  const float px  = (float)w * (351.0f / 21.0f);
  const float py  = (float)h * (127.0f / 7.0f);
  const float ppx = (px - u0) / g1;
  const float ppy = (py - v0) / g2;
  float pux = ppx, puy = ppy;
  #pragma unroll 1
  for (int it = 0; it < 20; ++it) {
    float r2 = pux*pux + puy*puy;
    float r4 = r2 * r2;
    float den = 1.0f + dk1*r2 + dk2*r4;
    float nx_ = (ppx - 2.0f*dp1*pux*puy - dp2*(r2 + 2.0f*pux*pux)) / den;
    float ny_ = (ppy - 2.0f*dp2*pux*puy - dp1*(r2 + 2.0f*puy*puy)) / den;
    pux = nx_; puy = ny_;
  }

  // ---- MEI sphere unprojection -> normalized ray (computed exactly once) ----
  float rxg, ryg, rzg;
  {
    float r2  = pux*pux + puy*puy;
    float aa  = r2 + 1.0f;
    float bb2 = 2.0f * xiv * r2;
    float cc  = xiv*xiv*r2 - 1.0f;
    float Zs  = (-bb2 + sqrtf(bb2*bb2 - 4.0f*aa*cc)) / (2.0f*aa);
    float rx  = pux*(Zs + xiv), ry = puy*(Zs + xiv), rz = Zs;
    float inv = 1.0f / sqrtf(rx*rx + ry*ry + rz*rz);
    rxg = rx*inv; ryg = ry*inv; rzg = rz*inv;
  }

  // ---- extrinsics ----
  const float R00 = rots[cam*9+0], R01 = rots[cam*9+1], R02 = rots[cam*9+2];
  const float R10 = rots[cam*9+3], R11 = rots[cam*9+4], R12 = rots[cam*9+5];
  const float R20 = rots[cam*9+6], R21 = rots[cam*9+7], R22 = rots[cam*9+8];
  const float tx = trans[cam*3+0], ty = trans[cam*3+1], tz = trans[cam*3+2];

  // ---- softmax over D=41 (lane holds d=lane and d=lane+32) ----
  const float* dl = depth_logits + (size_t)cam * (DDp*HH*WW) + h*WW + w;
  float x0 = (lane      < DDp) ? dl[(size_t)lane       * (HH*WW)] : -1e30f;
  float x1 = (lane + 32 < DDp) ? dl[(size_t)(lane + 32)* (HH*WW)] : -1e30f;
  float mv = fmaxf(x0, x1);
  #pragma unroll
  for (int off = 16; off > 0; off >>= 1) mv = fmaxf(mv, __shfl_xor(mv, off));
  float e0 = expf(x0 - mv);
  float e1 = expf(x1 - mv);
  float sv = e0 + e1;
  #pragma unroll
  for (int off = 16; off > 0; off >>= 1) sv += __shfl_xor(sv, off);
  const float pr0 = e0 / sv;   // prob for d = lane
  const float pr1 = e1 / sv;   // prob for d = lane+32 (0 for padded lanes)

  // ---- per-d voxel flat base (c added later), -1 if dropped ----
  auto vox = [&](int d) -> int {
    float dc = 4.0f + (float)d;              // depth candidate
    float pcx = rxg*dc, pcy = ryg*dc, pcz = rzg*dc;
    float ex = R00*pcx + R01*pcy + R02*pcz + tx;
    float ey = R10*pcx + R11*pcy + R12*pcz + ty;
    float ez = R20*pcx + R21*pcy + R22*pcz + tz;
    ex = -ex;                                 // * [-1, 1, 1]
    int ix = (int)floorf(ex*2.0f + 100.0f);   // floor((x+50)/0.5)
    int iy = (int)floorf(ey*2.0f + 100.0f);
    int iz = (int)floorf((ez + 10.0f) * (1.0f/20.0f));
    bool kept = (ix >= 0) && (ix < NXY) && (iy >= 0) && (iy < NXY) && (iz == 0);
    return kept ? (b*OUT_PER_B + ix*NXY + iy) : -1;
  };
  const int vb0 = (lane      < DDp) ? vox(lane)      : -1;
  const int vb1 = (lane + 32 < DDp) ? vox(lane + 32) : -1;

  // ---- features (lane holds c=lane and c=lane+32) ----
  const float* fp = feats + (size_t)cam * (CC*HH*WW) + h*WW + w;
  const float f0 = fp[(size_t)lane        * (HH*WW)];
  const float f1 = fp[(size_t)(lane + 32) * (HH*WW)];

  // ---- hoisted cross-lane operand variants ----
  const int lo   = lane & 15;
  const int hi8  = (lane >> 4) << 3;                 // 0 or 8
  const float pm_arr[3] = { pr0, __shfl(pr0, lane + 16), pr1 };
  const float fn_arr[4] = { f0, __shfl(f0, lane + 16), f1, __shfl(f1, lane + 16) };

  // voxel base per (d-tile, result VGPR r): this lane's row is m = r + hi8
  int vrow[3][8];
  #pragma unroll
  for (int r = 0; r < 8; ++r) {
    vrow[0][r] = __shfl(vb0,      r + hi8);          // d =      r + hi8
    vrow[1][r] = __shfl(vb0, 16 + r + hi8);          // d = 16 + r + hi8
    vrow[2][r] = __shfl(vb1,      r + hi8);          // d = 32 + r + hi8
  }

  // ---- lift (12 x WMMA f32 outer-product tiles) + atomic splat ----
  #pragma unroll
  for (int dt = 0; dt < 3; ++dt) {        // d tiles: 0-15, 16-31, 32-47(pad)
    #pragma unroll
    for (int ct = 0; ct < 4; ++ct) {      // c tiles: 4 x 16
      v8f tile = outer_wmma(pm_arr[dt], fn_arr[ct], lane);
      #pragma unroll
      for (int r = 0; r < 8; ++r) {
        const int vb = vrow[dt][r];
        if (vb >= 0) {
          const int c = ct*16 + lo;
          atomicAdd(out + (size_t)vb + (size_t)c * CSTRIDE, tile[r]);
        }
      }
    }
  }
}

extern "C" void kernel_launch(void* const* d_in, const int* in_sizes, int n_in,
                              void* d_out, int out_size, void* d_ws, size_t ws_size,
                              hipStream_t stream) {
  (void)in_sizes; (void)n_in; (void)d_ws; (void)ws_size; (void)out_size;
  const float* depth_logits = (const float*)d_in[0];
  const float* feats        = (const float*)d_in[1];
  const float* rots         = (const float*)d_in[2];
  const float* trans        = (const float*)d_in[3];
  const float* Kc           = (const float*)d_in[4];
  const float* Dc           = (const float*)d_in[5];
  const float* xiP          = (const float*)d_in[6];
  float* out = (float*)d_out;

  // 1) zero the BEV accumulator (8*64*200*200 floats = 5,120,000 float4)
  const int n4 = (BB * CC * CSTRIDE) / 4;
  lss_zero_f4<<<(n4 + 255) / 256, 256, 0, stream>>>((float4*)out, n4);

  // 2) one wave32 per pixel: 8448 pixels / 8 waves-per-block = 1056 blocks
  const int npix = BB * NNC * HH * WW;
  lss_lift_splat<<<(npix + 7) / 8, 256, 0, stream>>>(
      depth_logits, feats, rots, trans, Kc, Dc, xiP, out);
}